// AccessControlHead_81587198755030
// MI455X (gfx1250) — compile-verified
//
#include <hip/hip_runtime.h>
#include <hip/hip_bf16.h>

#define MODIFIER_COL 11
#define OWNER_COL    24

typedef float v2f __attribute__((ext_vector_type(2)));
typedef float v8f __attribute__((ext_vector_type(8)));

// ---------------------------------------------------------------------------
// Kernel 1: zero the workspace accumulators (cnt[G], s0[G], s1[G])
// ---------------------------------------------------------------------------
__global__ void ach_init_ws(float* __restrict__ ws, int n) {
    int i = blockIdx.x * blockDim.x + threadIdx.x;
    if (i < n) ws[i] = 0.0f;
}

// ---------------------------------------------------------------------------
// Kernel 2: segment sum of columns 11/24 + counts, exploiting sorted batch.
// Each thread owns RUN consecutive nodes; accumulates while the graph id is
// unchanged and flushes one atomicAdd triple per run boundary (~1.02 runs per
// thread on average for 488 nodes/graph). Only 2 dwords are touched per
// 512-byte row, both inside the row's first 128B cacheline -> ~128MB traffic.
// ---------------------------------------------------------------------------
#define RUN 8

__global__ __launch_bounds__(256) void ach_segsum(
    const float* __restrict__ nf,     // N x F row-major
    const int*   __restrict__ batch,  // N, sorted
    float* __restrict__ cnt,
    float* __restrict__ s0,
    float* __restrict__ s1,
    int N, int F) {
    long long start = (long long)(blockIdx.x * blockDim.x + threadIdx.x) * RUN;
    if (start >= N) return;
    long long end = start + RUN;
    if (end > N) end = N;

    int   g  = batch[start];
    float c  = 0.0f, a0 = 0.0f, a1 = 0.0f;
    for (long long i = start; i < end; ++i) {
        int gi = batch[i];
        if (gi != g) {
            atomicAdd(&cnt[g], c);
            atomicAdd(&s0[g], a0);
            atomicAdd(&s1[g], a1);
            g = gi; c = 0.0f; a0 = 0.0f; a1 = 0.0f;
        }
        const float* row = nf + i * (long long)F;
        a0 += row[MODIFIER_COL];
        a1 += row[OWNER_COL];
        c  += 1.0f;
    }
    atomicAdd(&cnt[g], c);
    atomicAdd(&s0[g], a0);
    atomicAdd(&s1[g], a1);
}

// ---------------------------------------------------------------------------
// Kernel 3: per-graph MLP using V_WMMA_F32_16X16X4_F32.
// One wave32 handles a tile of 16 graphs:
//   A (16x4, K=2 used): lanes 0-15 -> a = {af0, af1}; lanes 16-31 -> 0 (K=2,3 pad)
//   B (4x16):           lanes 0-15 -> b = {W1[n][0], W1[n][1]}; lanes 16-31 -> 0
//   Two WMMAs cover hidden units n=0..15 and n=16..31.
// C/D layout: VGPR v = row v (lanes 0-15) / row v+8 (lanes 16-31), col = lane%16.
// Layer 2 = per-lane fma with W2[col] then shuffle-reduce across 16-lane groups.
// EXEC is full around the WMMAs (tile count divides the wave count exactly and
// pre-WMMA divergence only predicates loads).
// ---------------------------------------------------------------------------
__global__ __launch_bounds__(256) void ach_mlp_wmma(
    const float* __restrict__ cnt,
    const float* __restrict__ s0,
    const float* __restrict__ s1,
    const float* __restrict__ W1,   // (32,2) row-major
    const float* __restrict__ b1,   // (32,)
    const float* __restrict__ W2,   // (1,32)
    const float* __restrict__ b2,   // (1,)
    float* __restrict__ out,        // (G,)
    int G) {
    const int lane = threadIdx.x & 31;
    const int wave = threadIdx.x >> 5;
    const int tile = blockIdx.x * (blockDim.x >> 5) + wave;
    const int g0   = tile * 16;
    if (g0 >= G) return;  // wave-uniform

    const int  half = lane & 15;
    const bool lo   = lane < 16;

    // --- A fragment: access_features = 1 - segment_mean -------------------
    v2f a; a.x = 0.0f; a.y = 0.0f;
    if (lo) {
        int   g = g0 + half;
        float d = fmaxf(cnt[g], 1.0f);
        a.x = 1.0f - s0[g] / d;
        a.y = 1.0f - s1[g] / d;
    }

    // --- B fragments: W1^T columns (hidden units 0-15 and 16-31) ----------
    v2f bA; bA.x = 0.0f; bA.y = 0.0f;
    v2f bB; bB.x = 0.0f; bB.y = 0.0f;
    if (lo) {
        bA.x = W1[half * 2 + 0];        bA.y = W1[half * 2 + 1];
        bB.x = W1[(half + 16) * 2 + 0]; bB.y = W1[(half + 16) * 2 + 1];
    }

    v8f c0 = {};
    v8f c1 = {};
    c0 = __builtin_amdgcn_wmma_f32_16x16x4_f32(
        /*neg_a=*/false, a, /*neg_b=*/false, bA,
        /*c_mod=*/(short)0, c0, /*reuse_a=*/false, /*reuse_b=*/false);
    c1 = __builtin_amdgcn_wmma_f32_16x16x4_f32(
        /*neg_a=*/false, a, /*neg_b=*/false, bB,
        /*c_mod=*/(short)0, c1, /*reuse_a=*/false, /*reuse_b=*/false);

    // --- bias + ReLU + layer-2 partials (col n0 for c0, n1 for c1) --------
    const int   n0 = half, n1 = half + 16;
    const float bias0 = b1[n0], bias1 = b1[n1];
    const float w20   = W2[n0], w21   = W2[n1];

    float p[8];
#pragma unroll
    for (int v = 0; v < 8; ++v) {
        float h0 = fmaxf(c0[v] + bias0, 0.0f);
        float h1 = fmaxf(c1[v] + bias1, 0.0f);
        p[v] = h0 * w20 + h1 * w21;
    }

    // --- reduce across each 16-lane group (rows 0-7 | rows 8-15) ----------
#pragma unroll
    for (int m = 8; m >= 1; m >>= 1) {
#pragma unroll
        for (int v = 0; v < 8; ++v) p[v] += __shfl_xor(p[v], m, 16);
    }

    // --- sigmoid + empty-graph mask; lanes 0 and 16 write 8 rows each -----
    if (half == 0) {
        const int   rowBase = lo ? 0 : 8;
        const float bias2   = b2[0];
#pragma unroll
        for (int v = 0; v < 8; ++v) {
            int   g     = g0 + rowBase + v;
            float z     = p[v] + bias2;
            float score = 1.0f / (1.0f + __expf(-z));
            out[g] = (cnt[g] > 0.0f) ? score : 0.0f;
        }
    }
}

// ---------------------------------------------------------------------------
// Host-side launcher
// Inputs (setup_inputs order):
//   0: node_features (N*F f32)   1: batch (N int)        2: graph_embedding (unused)
//   3: W1 (32*2 f32)             4: b1 (32 f32)          5: W2 (1*32 f32)
//   6: b2 (1 f32)                7: num_graphs (scalar)
// Output: G f32 scores. G == out_size.
// ---------------------------------------------------------------------------
extern "C" void kernel_launch(void* const* d_in, const int* in_sizes, int n_in,
                              void* d_out, int out_size, void* d_ws, size_t ws_size,
                              hipStream_t stream) {
    const float* nf    = (const float*)d_in[0];
    const int*   batch = (const int*)d_in[1];
    const float* W1    = (const float*)d_in[3];
    const float* b1    = (const float*)d_in[4];
    const float* W2    = (const float*)d_in[5];
    const float* b2    = (const float*)d_in[6];
    float*       out   = (float*)d_out;

    const int N = in_sizes[1];
    const int F = in_sizes[0] / N;   // 128
    const int G = out_size;          // 2048

    float* cnt = (float*)d_ws;
    float* s0  = cnt + G;
    float* s1  = s0 + G;

    // 1) zero accumulators
    {
        int n   = 3 * G;
        int blk = 256;
        ach_init_ws<<<(n + blk - 1) / blk, blk, 0, stream>>>(cnt, n);
    }
    // 2) segment sums
    {
        long long threads = ((long long)N + RUN - 1) / RUN;
        int blk = 256;
        int grd = (int)((threads + blk - 1) / blk);
        ach_segsum<<<grd, blk, 0, stream>>>(nf, batch, cnt, s0, s1, N, F);
    }
    // 3) WMMA MLP: 16 graphs per wave, 8 waves per block
    {
        int tiles = (G + 15) / 16;               // 128
        int blk   = 256;                         // 8 waves
        int grd   = (tiles + 7) / 8;             // 16
        ach_mlp_wmma<<<grd, blk, 0, stream>>>(cnt, s0, s1, W1, b1, W2, b2, out, G);
    }
}